// EncoderBlock_87737591922976
// MI455X (gfx1250) — compile-verified
//
#include <hip/hip_runtime.h>

#define D 64

typedef __attribute__((ext_vector_type(2))) float v2f;
typedef __attribute__((ext_vector_type(8))) float v8f;

// ---------------------------------------------------------------------------
// zero a float buffer (float4 granularity; counts here are multiples of 4)
// ---------------------------------------------------------------------------
__global__ __launch_bounds__(256) void zero_kernel(float4* __restrict__ p, long n4) {
    long i = (long)blockIdx.x * blockDim.x + threadIdx.x;
    if (i < n4) p[i] = make_float4(0.f, 0.f, 0.f, 0.f);
}

// ---------------------------------------------------------------------------
// deg[dst[e]] += ew[e]
// ---------------------------------------------------------------------------
__global__ __launch_bounds__(256) void deg_kernel(const float* __restrict__ ew,
                                                  const int* __restrict__ dst,
                                                  float* __restrict__ deg, int E) {
    int e = blockIdx.x * blockDim.x + threadIdx.x;
    if (e < E) unsafeAtomicAdd(&deg[dst[e]], ew[e]);
}

// deg -> dis in place: dis = deg>0 ? rsqrt(deg) : 0
__global__ __launch_bounds__(256) void dis_kernel(float* __restrict__ deg, int n) {
    int i = blockIdx.x * blockDim.x + threadIdx.x;
    if (i < n) {
        float d = deg[i];
        deg[i] = d > 0.f ? rsqrtf(d) : 0.f;
    }
}

// norm[e] = dis[src]*ew*dis[dst]
__global__ __launch_bounds__(256) void norm_kernel(const float* __restrict__ dis,
                                                   const float* __restrict__ ew,
                                                   const int* __restrict__ src,
                                                   const int* __restrict__ dst,
                                                   float* __restrict__ norm, int E) {
    int e = blockIdx.x * blockDim.x + threadIdx.x;
    if (e < E) norm[e] = dis[src[e]] * ew[e] * dis[dst[e]];
}

// ---------------------------------------------------------------------------
// GEMM: Y[n,64] = X[n,64] @ W[64,64] (+bias), f32 WMMA 16x16x4.
// Block = 256 threads = 8 waves; block computes 32 rows x 64 cols.
// Wave w: row-tile = (w>>2)*16, col-tile = (w&3)*16.
// Safe for X == Y (X rows fully staged in LDS before any store).
// ---------------------------------------------------------------------------
__global__ __launch_bounds__(256) void gemm64_wmma(const float* __restrict__ X,
                                                   const float* __restrict__ W,
                                                   const float* __restrict__ bias,
                                                   float* __restrict__ Y, int n) {
    __shared__ float Ws[64 * 64];   // 16 KB
    __shared__ float Xs[32 * 64];   //  8 KB
    const int t = threadIdx.x;
    const int blockRow = blockIdx.x * 32;

    // cooperative load of W (1024 float4) and X tile (512 float4)
    const float4* W4 = (const float4*)W;
    float4* Ws4 = (float4*)Ws;
    #pragma unroll
    for (int i = t; i < 1024; i += 256) Ws4[i] = W4[i];

    const float4* X4 = (const float4*)(X + (long)blockRow * D);
    float4* Xs4 = (float4*)Xs;
    for (int i = t; i < 512; i += 256) {
        int row = blockRow + (i >> 4);               // 16 float4 per row
        Xs4[i] = (row < n) ? X4[i] : make_float4(0.f, 0.f, 0.f, 0.f);
    }
    __syncthreads();

    const int wave = t >> 5;
    const int lane = t & 31;
    const int rowTile = (wave >> 2) * 16;            // 0 or 16
    const int colBase = (wave & 3) * 16;
    const int m    = lane & 15;
    const int koff = (lane >> 4) << 1;               // lanes 16..31 handle K+2,K+3

    v8f acc = {};
    #pragma unroll
    for (int k = 0; k < D; k += 4) {
        const int kk = k + koff;
        v2f A, B;
        A.x = Xs[(rowTile + m) * D + kk];
        A.y = Xs[(rowTile + m) * D + kk + 1];
        B.x = Ws[kk * D + colBase + m];
        B.y = Ws[(kk + 1) * D + colBase + m];
        acc = __builtin_amdgcn_wmma_f32_16x16x4_f32(false, A, false, B,
                                                    (short)0, acc, false, false);
    }

    const int col = colBase + m;
    const float bv = bias ? bias[col] : 0.f;
    const int rbase = blockRow + rowTile + ((lane >> 4) << 3); // +8 for hi half
    #pragma unroll
    for (int v = 0; v < 8; ++v) {
        int row = rbase + v;
        if (row < n) Y[(long)row * D + col] = acc[v] + bv;
    }
}

// ---------------------------------------------------------------------------
// scatter: agg[dst[e]] += h[src[e]] * norm[e]   (float4 loads, f32 atomics)
// one thread per (edge, 4-elem chunk): E*16 threads
// ---------------------------------------------------------------------------
__global__ __launch_bounds__(256) void scatter_kernel(const float* __restrict__ h,
                                                      const int* __restrict__ src,
                                                      const int* __restrict__ dst,
                                                      const float* __restrict__ norm,
                                                      float* __restrict__ agg, int E) {
    long tid = (long)blockIdx.x * blockDim.x + threadIdx.x;
    int e = (int)(tid >> 4);
    int q = (int)(tid & 15);
    if (e >= E) return;
    float nv = norm[e];
    const float4* hp = (const float4*)(h + (long)src[e] * D);
    float4 v = hp[q];
    float* a = agg + (long)dst[e] * D + q * 4;
    unsafeAtomicAdd(a + 0, v.x * nv);
    unsafeAtomicAdd(a + 1, v.y * nv);
    unsafeAtomicAdd(a + 2, v.z * nv);
    unsafeAtomicAdd(a + 3, v.w * nv);
}

// ---------------------------------------------------------------------------
// per node: t = in + bias; LayerNorm(t)*lnw+lnb; exact GELU.  one wave/node.
// ---------------------------------------------------------------------------
__global__ __launch_bounds__(256) void ln_gelu_kernel(const float* __restrict__ in,
                                                      const float* __restrict__ bias,
                                                      const float* __restrict__ lnw,
                                                      const float* __restrict__ lnb,
                                                      float* __restrict__ out, int n) {
    int node = (int)(((long)blockIdx.x * blockDim.x + threadIdx.x) >> 5);
    int lane = threadIdx.x & 31;
    if (node >= n) return;
    const float* p = in + (long)node * D;
    float x0 = p[lane], x1 = p[lane + 32];
    if (bias) { x0 += bias[lane]; x1 += bias[lane + 32]; }
    float s = x0 + x1;
    #pragma unroll
    for (int off = 16; off; off >>= 1) s += __shfl_xor(s, off, 32);
    float mu = s * (1.0f / D);
    float d0 = x0 - mu, d1 = x1 - mu;
    float v = d0 * d0 + d1 * d1;
    #pragma unroll
    for (int off = 16; off; off >>= 1) v += __shfl_xor(v, off, 32);
    float rstd = rsqrtf(v * (1.0f / D) + 1e-5f);
    float h0 = d0 * rstd * lnw[lane] + lnb[lane];
    float h1 = d1 * rstd * lnw[lane + 32] + lnb[lane + 32];
    h0 = 0.5f * h0 * (1.0f + erff(h0 * 0.70710678118654752f));
    h1 = 0.5f * h1 * (1.0f + erff(h1 * 0.70710678118654752f));
    float* q = out + (long)node * D;
    q[lane] = h0;
    q[lane + 32] = h1;
}

// ---------------------------------------------------------------------------
// out = (conv_out + identity) / max(||.||2, 1e-8).  one wave/node.
// ---------------------------------------------------------------------------
__global__ __launch_bounds__(256) void finalize_kernel(const float* __restrict__ conv,
                                                       const float* __restrict__ ident,
                                                       float* __restrict__ out, int n) {
    int node = (int)(((long)blockIdx.x * blockDim.x + threadIdx.x) >> 5);
    int lane = threadIdx.x & 31;
    if (node >= n) return;
    long base = (long)node * D;
    float x0 = conv[base + lane] + ident[base + lane];
    float x1 = conv[base + lane + 32] + ident[base + lane + 32];
    float ss = x0 * x0 + x1 * x1;
    #pragma unroll
    for (int off = 16; off; off >>= 1) ss += __shfl_xor(ss, off, 32);
    float inv = 1.0f / fmaxf(sqrtf(ss), 1e-8f);
    out[base + lane] = x0 * inv;
    out[base + lane + 32] = x1 * inv;
}

// ---------------------------------------------------------------------------
extern "C" void kernel_launch(void* const* d_in, const int* in_sizes, int n_in,
                              void* d_out, int out_size, void* d_ws, size_t ws_size,
                              hipStream_t stream) {
    const float* x        = (const float*)d_in[0];
    const int*   eidx     = (const int*)d_in[1];
    const float* ew       = (const float*)d_in[2];
    const float* conv_w   = (const float*)d_in[3];
    const float* conv_b   = (const float*)d_in[4];
    const float* ln_w     = (const float*)d_in[5];
    const float* ln_b     = (const float*)d_in[6];
    const float* res_w    = (const float*)d_in[7];
    const float* res_b    = (const float*)d_in[8];
    const float* res_ln_w = (const float*)d_in[9];
    const float* res_ln_b = (const float*)d_in[10];

    const int N = in_sizes[0] / D;
    const int E = in_sizes[2];
    const int n_convs = in_sizes[3] / (D * D);
    const int* src = eidx;
    const int* dst = eidx + E;

    float* wsf      = (float*)d_ws;
    float* deg      = wsf;                       // N  (becomes dis in place)
    float* norm     = deg + N;                   // E
    float* identity = norm + E;                  // N*D
    float* bufH     = identity + (long)N * D;    // N*D  (h, then layer output)
    float* bufAgg   = bufH + (long)N * D;        // N*D

    const int TB = 256;
    const int gN    = (N + TB - 1) / TB;
    const int gE    = (E + TB - 1) / TB;
    const int gND4  = (int)(((long)N * D / 4 + TB - 1) / TB);
    const int gGemm = (N + 31) / 32;
    const int gWaveN = (N * 32 + TB - 1) / TB;          // one wave per node
    const int gScat = (int)(((long)E * 16 + TB - 1) / TB);

    // --- degree / normalization ---
    zero_kernel<<<(N / 4 + TB - 1) / TB, TB, 0, stream>>>((float4*)deg, N / 4);
    deg_kernel<<<gE, TB, 0, stream>>>(ew, dst, deg, E);
    dis_kernel<<<gN, TB, 0, stream>>>(deg, N);
    norm_kernel<<<gE, TB, 0, stream>>>(deg, ew, src, dst, norm, E);

    // --- residual branch: gelu(LN(x @ res_w + res_b)) -> identity ---
    gemm64_wmma<<<gGemm, TB, 0, stream>>>(x, res_w, res_b, bufH, N);
    ln_gelu_kernel<<<gWaveN, TB, 0, stream>>>(bufH, nullptr, res_ln_w, res_ln_b,
                                              identity, N);

    // --- conv stack ---
    for (int i = 0; i < n_convs; ++i) {
        const float* in = (i == 0) ? x : bufH;   // in-place-safe GEMM
        gemm64_wmma<<<gGemm, TB, 0, stream>>>(in, conv_w + (long)i * D * D,
                                              nullptr, bufH, N);
        zero_kernel<<<gND4, TB, 0, stream>>>((float4*)bufAgg, (long)N * D / 4);
        scatter_kernel<<<gScat, TB, 0, stream>>>(bufH, src, dst, norm, bufAgg, E);
        ln_gelu_kernel<<<gWaveN, TB, 0, stream>>>(bufAgg, conv_b + (long)i * D,
                                                  ln_w + (long)i * D,
                                                  ln_b + (long)i * D, bufH, N);
    }

    // --- residual add + L2 normalize ---
    finalize_kernel<<<gWaveN, TB, 0, stream>>>(bufH, identity, (float*)d_out, N);
}